// RoIExtract_72060961292715
// MI455X (gfx1250) — compile-verified
//
#include <hip/hip_runtime.h>
#include <hip/hip_bf16.h>
#include <cstdint>

typedef float v4f __attribute__((ext_vector_type(4)));
typedef unsigned int v4u __attribute__((ext_vector_type(4)));
typedef int v8i __attribute__((ext_vector_type(8)));
typedef int v4i __attribute__((ext_vector_type(4)));

#define ROI_C   32
#define ROI_H   8
#define ROI_W   128
#define CHUNK_FLOATS 1024            /* one (roi, c) slab: 8 rows x 128 cols, contiguous */
#define CHUNK_BYTES  4096
#define CHUNKS_PER_BLOCK 16          /* one block = half the channels of one sorted ROI */

// ---------------------------------------------------------------------------
// Kernel 1: widths[i] = min(ceil(8 * boxes[i,2] / boxes[i,3]), 128)
// ((8*(x/4))/(y/4) == (8x)/y exactly: power-of-two scalings are exact, one
//  rounded IEEE division either way; hipcc emits the correctly-rounded
//  v_div_scale/v_rcp sequence, matching jnp bit-for-bit)
// ---------------------------------------------------------------------------
__global__ void roi_widths_kernel(const float* __restrict__ boxes,
                                  int* __restrict__ widths, int N) {
    int i = blockIdx.x * blockDim.x + threadIdx.x;
    if (i < N) {
        float w = boxes[i * 4 + 2];
        float h = boxes[i * 4 + 3];
        int wi = (int)ceilf(8.0f * w / h);
        widths[i] = wi < ROI_W ? wi : ROI_W;
    }
}

// ---------------------------------------------------------------------------
// Kernel 2: stable descending counting sort over width values 1..128.
// Single block, 128 threads; thread t owns width value v = 128 - t.
// Each bucket filled by one thread scanning in index order -> stable,
// deterministic. Also emits float-typed tail outputs (widths[order], order).
// ---------------------------------------------------------------------------
__global__ void roi_sort_kernel(const int* __restrict__ widths,
                                int* __restrict__ order,
                                int* __restrict__ swidths,
                                float* __restrict__ out_tail, int N) {
    __shared__ int counts[128];
    __shared__ int bases[129];
    const int t = threadIdx.x;        // 0..127
    const int v = 128 - t;            // descending width value

    int cnt = 0;
    for (int i = 0; i < N; ++i) cnt += (widths[i] == v);
    counts[t] = cnt;
    __syncthreads();
    if (t == 0) {
        int acc = 0;
        for (int q = 0; q < 128; ++q) { bases[q] = acc; acc += counts[q]; }
        bases[128] = acc;
    }
    __syncthreads();
    int pos = bases[t];
    for (int i = 0; i < N; ++i) {
        if (widths[i] == v) {
            order[pos]   = i;
            swidths[pos] = v;
            out_tail[pos]     = (float)v;   // widths[order]
            out_tail[N + pos] = (float)i;   // order
            ++pos;
        }
    }
}

// ---------------------------------------------------------------------------
// Kernel 3: main copy via Tensor Data Mover.
// Block b: sorted slot k = b>>1 (block-constant!), channels c0..c0+15.
// Chunk j = channel c0+j: 4 KB contiguous slab at (mapping[order[k]]*32+c)<<16
// bytes. Wave 0 issues one TENSOR_LOAD_TO_LDS per chunk (1-D 4 KB tile),
// double-buffered on TENSORcnt; all waves then ds_load -> *mask -> 128-bit
// non-temporal global store.
// ---------------------------------------------------------------------------
__global__ void __launch_bounds__(256)
roi_copy_kernel(const float* __restrict__ fm,
                const int* __restrict__ mapping,
                const int* __restrict__ order,
                const int* __restrict__ swidths,
                float* __restrict__ out, int N) {
    __shared__ float buf[2][CHUNK_FLOATS];

    const int t   = threadIdx.x;      // 0..255
    const int o   = t * 4;            // float offset within chunk
    const int wx0 = o & (ROI_W - 1);  // column of first element (wave == one row)

    // ---- block-uniform metadata (hoisted: one scalar load chain total) ----
    const long long base_ch = (long long)blockIdx.x * CHUNKS_PER_BLOCK;
    const int k     = (int)(base_ch >> 5);        // sorted ROI slot (constant)
    const int c0    = (int)(base_ch & 31);        // first channel of this block
    const int src   = order[k];
    const int b     = mapping[src];
    const int width = swidths[k];

    const unsigned long long gbase =
        (unsigned long long)(const void*)fm +
        ((unsigned long long)(b * ROI_C + c0) << 16);   // bytes, + j<<16 per chunk

    // precomputed 0/1 multiplier (reference does rois * mask)
    v4f mmul;
#pragma unroll
    for (int q = 0; q < 4; ++q) mmul[q] = (wx0 + q < width) ? 1.0f : 0.0f;

    const unsigned lds_base[2] = {
        (unsigned)(uintptr_t)(void*)&buf[0][0],
        (unsigned)(uintptr_t)(void*)&buf[1][0]
    };

    // ---- TDM descriptor group1 (block-invariant): 1-D tile, 4-byte elems,
    //      tensor_dim0 = tile_dim0 = 1024, stride 1024, no pad/iterate ------
    v8i g1;
    g1[0] = 0x00020000;                 // workgroup_mask=0 | data_size=2 (4B)
    g1[1] = (int)(CHUNK_FLOATS << 16);  // tensor_dim0[15:0] @ bits 63:48
    g1[2] = (int)(1u << 16);            // tensor_dim1 = 1   @ bits 111:80
    g1[3] = (int)(CHUNK_FLOATS << 16);  // tile_dim0  = 1024 @ bits 127:112
    g1[4] = 1;                          // tile_dim1  = 1
    g1[5] = CHUNK_FLOATS;               // tensor_dim0_stride = 1024
    g1[6] = 0;
    g1[7] = 0;
    const v4i gz4 = {0, 0, 0, 0};               // D# groups 2/3 unused (<=2D tensor)
    const v8i gz8 = {0, 0, 0, 0, 0, 0, 0, 0};   // 6-arg builtin: extra group, zeroed

    const bool issuer = (t < 32);       // wave 0 drives the DMA engine

    auto issue_tdm = [&](int j) {
        unsigned long long ga = gbase + ((unsigned long long)j << 16);
        v4u g0;
        g0[0] = 1u;                                   // count=1, user mode
        g0[1] = lds_base[j & 1];                      // lds_addr
        g0[2] = (unsigned)(ga & 0xFFFFFFFFu);         // global_addr[31:0]
        g0[3] = (unsigned)((ga >> 32) & 0x01FFFFFFu)  // global_addr[56:32]
              | (2u << 30);                           // type = 2 (image)
        // clang-23 / therock-10.0 form: 6 args (extra int32x8 before cpol)
        __builtin_amdgcn_tensor_load_to_lds(g0, g1, gz4, gz4, gz8, 0);
    };

    if (issuer) issue_tdm(0);

    for (int j = 0; j < CHUNKS_PER_BLOCK; ++j) {
        if (issuer) {
            if (j + 1 < CHUNKS_PER_BLOCK) {
                issue_tdm(j + 1);                         // prefetch next buffer
                __builtin_amdgcn_s_wait_tensorcnt(1);     // chunk j landed (in-order)
            } else {
                __builtin_amdgcn_s_wait_tensorcnt(0);
            }
        }
        __syncthreads();                                  // DMA data visible to all waves

        v4f val = *(const v4f*)(&buf[j & 1][o]);          // ds_load_b128
        val *= mmul;                                      // column mask

        // 512 MB write stream: non-temporal so it doesn't evict the 4 MB hot
        // source set from L2 (each source byte re-read ~128x across ROIs).
        __builtin_nontemporal_store(
            val, (v4f*)(out + (((size_t)base_ch + j) << 10) + o));

        __syncthreads();    // buffer (j&1) drained before refill at j+2
    }
}

// ---------------------------------------------------------------------------
extern "C" void kernel_launch(void* const* d_in, const int* in_sizes, int n_in,
                              void* d_out, int out_size, void* d_ws, size_t ws_size,
                              hipStream_t stream) {
    const float* fm      = (const float*)d_in[0];   // (32,32,128,128) f32
    const float* boxes   = (const float*)d_in[1];   // (N,4) f32
    const int*   mapping = (const int*)d_in[2];     // (N,) i32
    float*       out     = (float*)d_out;
    const int N = in_sizes[2];

    int* ws       = (int*)d_ws;
    int* widths   = ws;             // N ints
    int* order    = ws + N;         // N ints
    int* swidths  = ws + 2 * N;     // N ints

    roi_widths_kernel<<<(N + 255) / 256, 256, 0, stream>>>(boxes, widths, N);

    float* out_tail = out + (size_t)N * (ROI_C * ROI_H * ROI_W);
    roi_sort_kernel<<<1, 128, 0, stream>>>(widths, order, swidths, out_tail, N);

    long long total_chunks = (long long)N * ROI_C;
    int blocks = (int)(total_chunks / CHUNKS_PER_BLOCK);
    roi_copy_kernel<<<blocks, 256, 0, stream>>>(fm, mapping, order, swidths, out, N);
}